// ANIRepresentation_47553877901498
// MI455X (gfx1250) — compile-verified
//
#include <hip/hip_runtime.h>
#include <hip/hip_bf16.h>
#include <stdint.h>

// ---------------------------------------------------------------------------
// ANI AEV (radial + angular) for MI455X / gfx1250, wave32.
// out row per atom: [ NSPECIES*NRBF = 112 radial | NPAIRS_SP*SUBLEN = 896 angular ]
//
// Atomic strategy: lanes compute their own pair/triple, wave-transpose the
// results through LDS, then all 32 lanes issue consecutive-address f32
// atomics into ONE 64B-aligned output row -> 1-2 L2 line-requests per atomic
// instruction instead of 32 scattered ones (16x fewer L2 atomic requests).
// ---------------------------------------------------------------------------

#define NSPECIES   7
#define NRBF       16
#define N_SHFA     8
#define N_SHFZ     4
#define SUBLEN     (N_SHFA * N_SHFZ)             // 32
#define NPAIRS_SP  (NSPECIES * (NSPECIES+1)/2)   // 28
#define RADLEN     (NSPECIES * NRBF)             // 112
#define OUTROW     (RADLEN + NPAIRS_SP * SUBLEN) // 1008

__device__ __constant__ const float kSHFR[NRBF] = {
  0.080000f, 0.106875f, 0.133750f, 0.160625f, 0.187500f, 0.214375f,
  0.241250f, 0.268125f, 0.295000f, 0.321875f, 0.348750f, 0.375625f,
  0.402500f, 0.429375f, 0.456250f, 0.483125f };

__device__ __constant__ const float kSHFA[N_SHFA] = {
  0.080000f, 0.113750f, 0.147500f, 0.181250f,
  0.215000f, 0.248750f, 0.282500f, 0.316250f };

// cos/sin of SHFZ = pi/8 + k*pi/4  (22.5, 67.5, 112.5, 157.5 deg)
__device__ __constant__ const float kCZ[N_SHFZ] = {
   0.92387953251f,  0.38268343236f, -0.38268343236f, -0.92387953251f };
__device__ __constant__ const float kSZ[N_SHFZ] = {
   0.38268343236f,  0.92387953251f,  0.92387953251f,  0.38268343236f };

constexpr float kRCR = 0.51f;
constexpr float kRCA = 0.35f;
constexpr float kPI  = 3.14159265358979323846f;
constexpr float kPiOverRcr = kPI / kRCR;
constexpr float kPiOverRca = kPI / kRCA;
constexpr float kEtaR = 1970.0f;
constexpr float kEtaA = 800.0f;

__device__ __forceinline__ float pow32f(float x) {
  x *= x; x *= x; x *= x; x *= x; x *= x;  // x^32 via 5 squarings
  return x;
}

// ---------------------------------------------------------------------------
// Kernel 0: zero-fill d_out using the CDNA5 async LDS->global store path.
// (verified last round: emits global_store_async_from_lds_b128)
// ---------------------------------------------------------------------------
__global__ __launch_bounds__(256) void zero_out_kernel(float* __restrict__ out,
                                                       long long nvec,   // # of 16B chunks
                                                       long long ntotal) // # of floats
{
  __shared__ uint32_t zbuf[1024]; // 4 KB of zeros
  for (int i = threadIdx.x; i < 1024; i += 256) zbuf[i] = 0u;
  __syncthreads();

  uint32_t lds_off = (uint32_t)(uintptr_t)(&zbuf[(threadIdx.x & 255u) * 4u]);

  const long long stride = (long long)gridDim.x * 256ll;
  for (long long v = (long long)blockIdx.x * 256ll + threadIdx.x; v < nvec; v += stride) {
    float* gptr = out + v * 4ll;
    asm volatile("global_store_async_from_lds_b128 %0, %1, off"
                 :: "v"(gptr), "v"(lds_off)
                 : "memory");
  }

  for (long long e = nvec * 4ll + (long long)blockIdx.x * 256ll + threadIdx.x;
       e < ntotal; e += stride) {
    out[e] = 0.0f;
  }

#if __has_builtin(__builtin_amdgcn_s_wait_asynccnt)
  __builtin_amdgcn_s_wait_asynccnt(0);
#endif
}

// ---------------------------------------------------------------------------
// Kernel 1: radial AEV. One lane per pair; wave-transposed atomics.
//   Row i (16 floats, 64B line) served by lanes 0-15, row j by lanes 16-31:
//   one atomic instruction per pair touching exactly two cachelines.
// ---------------------------------------------------------------------------
__global__ __launch_bounds__(256) void radial_kernel(
    const float* __restrict__ d_ij,
    const int*   __restrict__ pair_indices, // (2, npairs)
    const int*   __restrict__ species,
    float*       __restrict__ out,
    int npairs)
{
  __shared__ float tile[8][32][17];  // [wave][pair][rbf], pad 17 (gcd(17,64)=1)

  const int lane = threadIdx.x & 31;
  const int wave = threadIdx.x >> 5;
  const int pbase = blockIdx.x * 256 + wave * 32;
  const int p = pbase + lane;
  const bool valid = (p < npairs);

  int offi = 0, offj = 0;
  float vals[NRBF];
#pragma unroll
  for (int k = 0; k < NRBF; ++k) vals[k] = 0.0f;

  if (valid) {
    float d  = d_ij[p];
    float fc = (d <= kRCR) ? 0.5f * (__cosf(d * kPiOverRcr) + 1.0f) : 0.0f;
    float q  = 0.25f * fc;

    int i  = pair_indices[p];
    int j  = pair_indices[npairs + p];
    int si = species[i];
    int sj = species[j];
    offi = i * OUTROW + sj * NRBF;   // fits in int: out_size is int
    offj = j * OUTROW + si * NRBF;

#pragma unroll
    for (int k = 0; k < NRBF; ++k) {
      float dd = d - kSHFR[k];
      vals[k] = q * __expf(-kEtaR * dd * dd);
    }
  }

#pragma unroll
  for (int k = 0; k < NRBF; ++k) tile[wave][lane][k] = vals[k];
  __syncthreads();

  int nvalid = npairs - pbase;
  nvalid = nvalid < 0 ? 0 : (nvalid > 32 ? 32 : nvalid);

  const int col = lane & 15;
  for (int t = 0; t < nvalid; ++t) {
    int oi = __shfl(offi, t);
    int oj = __shfl(offj, t);
    int off = ((lane < 16) ? oi : oj) + col;
    atomicAdd(out + off, tile[wave][t][col]);
  }
}

// ---------------------------------------------------------------------------
// Kernel 2: angular AEV. One lane per triple; wave-transposed atomics.
// arccos eliminated: cos(theta - shfz) = c*CZ + s*SZ with s = sqrt(1-c^2)
// (exact; c is clamped to [-0.95, 0.95] by the 0.95 factor).
// Each output row is 32 floats starting 64B-aligned -> one atomic
// instruction per triple touching two cachelines.
// ---------------------------------------------------------------------------
__global__ __launch_bounds__(256) void angular_kernel(
    const float* __restrict__ r_ij,           // (npairs, 3)
    const int*   __restrict__ pair_indices,   // (2, npairs)
    const int*   __restrict__ species,
    const int*   __restrict__ close_idx,      // (nclose)
    const int*   __restrict__ central_atom,   // (ntrip)
    const int*   __restrict__ pair_index12,   // (2, ntrip)
    const int*   __restrict__ sign12,         // (2, ntrip)
    float*       __restrict__ out,
    int npairs, int ntrip)
{
  __shared__ float tile[8][32][33];  // [wave][triple][feature], pad 33

  const int lane = threadIdx.x & 31;
  const int wave = threadIdx.x >> 5;
  const int tbase = blockIdx.x * 256 + wave * 32;
  const int t = tbase + lane;
  const bool valid = (t < ntrip);

  int rowoff = 0;
  float vals[SUBLEN];
#pragma unroll
  for (int c = 0; c < SUBLEN; ++c) vals[c] = 0.0f;

  if (valid) {
    int p1 = pair_index12[t];
    int p2 = pair_index12[ntrip + t];
    int s1 = sign12[t];
    int s2 = sign12[ntrip + t];
    int cp1 = close_idx[p1];
    int cp2 = close_idx[p2];

    float g1 = (float)s1, g2 = (float)s2;
    float ax = r_ij[cp1 * 3 + 0] * g1;
    float ay = r_ij[cp1 * 3 + 1] * g1;
    float az = r_ij[cp1 * 3 + 2] * g1;
    float bx = r_ij[cp2 * 3 + 0] * g2;
    float by = r_ij[cp2 * 3 + 1] * g2;
    float bz = r_ij[cp2 * 3 + 2] * g2;

    // Species of the two non-central endpoints.
    int e1 = (s1 == 1) ? pair_indices[npairs + cp1] : pair_indices[cp1];
    int e2 = (s2 == 1) ? pair_indices[npairs + cp2] : pair_indices[cp2];
    int sp1 = species[e1];
    int sp2 = species[e2];

    float dd1 = ax * ax + ay * ay + az * az;
    float dd2 = bx * bx + by * by + bz * bz;
    float d1 = __builtin_sqrtf(dd1);
    float d2 = __builtin_sqrtf(dd2);

    float dot = ax * bx + ay * by + az * bz;
    float c   = __fdividef(0.95f * dot, d1 * d2);
    float s   = __builtin_sqrtf(fmaxf(1.0f - c * c, 0.0f));

    float fc1 = (d1 <= kRCA) ? 0.5f * (__cosf(d1 * kPiOverRca) + 1.0f) : 0.0f;
    float fc2 = (d2 <= kRCA) ? 0.5f * (__cosf(d2 * kPiOverRca) + 1.0f) : 0.0f;
    float scale = 2.0f * fc1 * fc2;

    float f1[N_SHFZ];
#pragma unroll
    for (int z = 0; z < N_SHFZ; ++z) {
      float base = 0.5f * (1.0f + c * kCZ[z] + s * kSZ[z]);
      f1[z] = pow32f(base);
    }

    float dm = 0.5f * (d1 + d2);
#pragma unroll
    for (int a = 0; a < N_SHFA; ++a) {
      float dd = dm - kSHFA[a];
      float fa = scale * __expf(-kEtaA * dd * dd);
#pragma unroll
      for (int z = 0; z < N_SHFZ; ++z) {
        vals[a * N_SHFZ + z] = fa * f1[z];
      }
    }

    // triu index of (sp1, sp2) in row-major upper triangle of 7x7.
    int r = min(sp1, sp2), cmx = max(sp1, sp2);
    int triu = r * NSPECIES - (r * (r - 1)) / 2 + (cmx - r);

    rowoff = central_atom[t] * OUTROW + RADLEN + triu * SUBLEN;
  }

#pragma unroll
  for (int c = 0; c < SUBLEN; ++c) tile[wave][lane][c] = vals[c];
  __syncthreads();

  int nvalid = ntrip - tbase;
  nvalid = nvalid < 0 ? 0 : (nvalid > 32 ? 32 : nvalid);

  for (int tt = 0; tt < nvalid; ++tt) {
    int off = __shfl(rowoff, tt) + lane;
    atomicAdd(out + off, tile[wave][tt][lane]);
  }
}

// ---------------------------------------------------------------------------
extern "C" void kernel_launch(void* const* d_in, const int* in_sizes, int n_in,
                              void* d_out, int out_size, void* d_ws, size_t ws_size,
                              hipStream_t stream) {
  const float* d_ij         = (const float*)d_in[0];
  const float* r_ij         = (const float*)d_in[1];
  const int*   pair_indices = (const int*)  d_in[2];
  const int*   species      = (const int*)  d_in[3];
  const int*   close_idx    = (const int*)  d_in[4];
  const int*   central      = (const int*)  d_in[5];
  const int*   pidx12       = (const int*)  d_in[6];
  const int*   sgn12        = (const int*)  d_in[7];
  (void)n_in; (void)d_ws; (void)ws_size;

  int npairs = in_sizes[0];      // d_ij is (npairs, 1)
  int ntrip  = in_sizes[5];      // central_atom_index length

  float* out = (float*)d_out;
  long long ntotal = (long long)out_size;
  long long nvec   = ntotal / 4ll;

  long long want = (nvec + 255ll) / 256ll;
  int zblocks = (int)(want < 1ll ? 1ll : (want > 4096ll ? 4096ll : want));
  zero_out_kernel<<<zblocks, 256, 0, stream>>>(out, nvec, ntotal);

  if (npairs > 0) {
    radial_kernel<<<(npairs + 255) / 256, 256, 0, stream>>>(
        d_ij, pair_indices, species, out, npairs);
  }
  if (ntrip > 0) {
    angular_kernel<<<(ntrip + 255) / 256, 256, 0, stream>>>(
        r_ij, pair_indices, species, close_idx, central, pidx12, sgn12,
        out, npairs, ntrip);
  }
}